// IndexedLinearLayer_45414984188344
// MI455X (gfx1250) — compile-verified
//
#include <hip/hip_runtime.h>

// Problem constants (match reference)
#define B_DIM    256
#define SIZE_IN  1024
#define SIZE_OUT 1024
#define NQ       16
#define OTILE    4      // outputs per block (indexed path)
#define KC       64     // K-chunk rows staged in LDS per iteration

typedef __attribute__((ext_vector_type(2))) float v2f;
typedef __attribute__((ext_vector_type(4))) float v4f;
typedef __attribute__((ext_vector_type(8))) float v8f;

#if __has_builtin(__builtin_amdgcn_global_load_async_to_lds_b128) && \
    __has_builtin(__builtin_amdgcn_s_wait_asynccnt)
#define USE_ASYNC_LDS 1
// Param 0 of the async builtin is int4 in AS(1) (global); LDS side is AS(3).
typedef int v4i __attribute__((vector_size(16)));
typedef __attribute__((address_space(1))) v4i* gv4i_p;   // global int4*
typedef __attribute__((address_space(3))) v4i* lv4i_p;   // LDS int4*
#endif

// D = A(16x4, f32) * B(4x16, f32) + C(16x16, f32), wave32.
// A layout: lane L holds row M=L&15; VGPR j holds K = 2*(L>>4)+j.
// B layout: lane L holds col N=L&15; VGPR j holds K = 2*(L>>4)+j.
// C/D layout: lane L holds col N=L&15; VGPR v holds row M = v + 8*(L>>4).
__device__ __forceinline__ v8f wmma_f32_4(v2f a, v2f b, v8f c) {
    return __builtin_amdgcn_wmma_f32_16x16x4_f32(
        /*neg_a=*/false, a, /*neg_b=*/false, b,
        /*c_mod=*/(short)0, c, /*reuse_a=*/false, /*reuse_b=*/false);
}

__global__ __launch_bounds__(256) void IndexedLinearLayer_wmma_kernel(
    const float* __restrict__ x,          // [B, SIZE_IN]
    const float* __restrict__ weights,    // [SIZE_OUT, SIZE_IN]
    const float* __restrict__ bias,       // [SIZE_OUT]
    const float* __restrict__ iw,         // [SIZE_OUT, SIZE_IN, NQ]
    const int*   __restrict__ indices,    // [B, SIZE_OUT]
    const int*   __restrict__ use_indices,// [1]
    float*       __restrict__ out)        // [B, SIZE_OUT]
{
    const int lane  = threadIdx.x & 31;
    const int wid   = threadIdx.x >> 5;
    const int gwave = blockIdx.x * (blockDim.x >> 5) + wid;

    const int nl = lane & 15;   // row (A) / col (B,C) within tile
    const int kh = lane >> 4;   // K-half select

    const v8f vzero = {0.f, 0.f, 0.f, 0.f, 0.f, 0.f, 0.f, 0.f};

    // Double-buffered LDS weight slab: 2 x (OTILE outputs x KC rows x NQ q) = 32 KB
    __shared__ float lsb[2][OTILE * KC * NQ];

    if (use_indices[0]) {
        // ---- Indexed path: C[m,q] = sum_k x[b0+m,k] * iw[o,k,q], then pick q=indices ----
        // Block: one o_group (4 outputs) x 8 batch tiles (one per wave).
        // 512 blocks: o_group = blockIdx>>1, b-half = blockIdx&1.
        const int o0 = (blockIdx.x >> 1) * OTILE;
        const int b0 = (((blockIdx.x & 1) << 3) + wid) * 16;

        v8f acc[OTILE];
        #pragma unroll
        for (int oo = 0; oo < OTILE; ++oo) acc[oo] = vzero;

        // A operand: this lane walks row (b0+nl), starting at column 2*kh
        const float* __restrict__ xrow = x + (size_t)(b0 + nl) * SIZE_IN + 2 * kh;

        // Staging assignment: 64 threads per output slab, 4 x b128 each per chunk.
        const int o_s = threadIdx.x >> 6;   // 0..3: which output this thread stages
        const int e   = threadIdx.x & 63;   // element group within the slab
        const size_t gslab = (size_t)(o0 + o_s) * SIZE_IN * NQ;  // float offset of o's slab
        const int    lslab = o_s * (KC * NQ);

        // WMMA inner chunk: KC/4 K-steps per output, B fragments from LDS (ds_read2)
        auto compute_chunk = [&](const float* lbuf, int k0) {
            #pragma unroll 4
            for (int k = 0; k < KC; k += 4) {
                v2f a;
                a.x = xrow[k0 + k];
                a.y = xrow[k0 + k + 1];
                #pragma unroll
                for (int oo = 0; oo < OTILE; ++oo) {
                    const float* lp = lbuf + (oo * KC + k + 2 * kh) * NQ + nl;
                    v2f b;
                    b.x = lp[0];    // row k+2*kh
                    b.y = lp[NQ];   // row k+2*kh+1  (64B apart -> ds_read2)
                    acc[oo] = wmma_f32_4(a, b, acc[oo]);
                }
            }
        };

#ifdef USE_ASYNC_LDS
        // Issue ASYNCcnt-tracked global->LDS b128 copies for chunk at k0 into buf.
        auto stage = [&](int k0, int buf) {
            const size_t gbase = gslab + (size_t)k0 * NQ;
            #pragma unroll
            for (int j = 0; j < 4; ++j) {
                const int f4 = (e + j * 64) * 4;   // float index of this b128
                __builtin_amdgcn_global_load_async_to_lds_b128(
                    (gv4i_p)(uintptr_t)(iw + gbase + f4),
                    (lv4i_p)(uint32_t)(uintptr_t)(&lsb[buf][lslab + f4]),
                    0, 0);
            }
        };

        // Prologue: fill buffer 0.
        stage(0, 0);
        __builtin_amdgcn_s_wait_asynccnt(0);
        __syncthreads();

        int cur = 0;
        for (int k0 = 0; k0 < SIZE_IN; k0 += KC, cur ^= 1) {
            // Overlap: issue next chunk's async copy into the other buffer.
            // (Safe: last iteration's barrier proved all waves finished reading it.)
            if (k0 + KC < SIZE_IN) stage(k0 + KC, cur ^ 1);
            compute_chunk(lsb[cur], k0);
            __builtin_amdgcn_s_wait_asynccnt(0);  // next buffer resident (this wave)
            __syncthreads();                      // ... and for all waves
        }
#else
        // Fallback: register-staged single buffer, two barriers per chunk.
        for (int k0 = 0; k0 < SIZE_IN; k0 += KC) {
            const size_t gbase = gslab + (size_t)k0 * NQ;
            v4f r[4];
            #pragma unroll
            for (int j = 0; j < 4; ++j)
                r[j] = *(const v4f*)(iw + gbase + (size_t)(e + j * 64) * 4);
            __syncthreads();    // previous chunk fully consumed
            #pragma unroll
            for (int j = 0; j < 4; ++j)
                *(v4f*)(&lsb[0][lslab + (e + j * 64) * 4]) = r[j];
            __syncthreads();    // staging visible to all waves
            compute_chunk(lsb[0], k0);
        }
#endif

        // ---- Selection: for row m, value C[m,q] sits in lane q + 16*(m>>3), VGPR (m&7). ----
        const int m = lane & 15;          // lanes 16..31 redundantly compute m of lanes 0..15
        #pragma unroll
        for (int oo = 0; oo < OTILE; ++oo) {
            const int o = o0 + oo;
            const int q = indices[(size_t)(b0 + m) * SIZE_OUT + o] & (NQ - 1);
            const int srcb = ((q & 15) + ((m >> 3) << 4)) << 2;  // byte index for ds_bpermute
            float val = 0.f;
            #pragma unroll
            for (int v = 0; v < 8; ++v) {
                int t = __builtin_amdgcn_ds_bpermute(srcb, __float_as_int(acc[oo][v]));
                if ((m & 7) == v) val = __int_as_float(t);
            }
            if (lane < 16)
                out[(size_t)(b0 + m) * SIZE_OUT + o] = val + bias[o];
        }
    } else {
        // ---- Plain path: out = x @ W.T + bias (16x16 output tiles) ----
        // 1024 tiles: o_tile = gwave & 63, b_tile = gwave >> 6 ; extra waves exit.
        if (gwave >= (B_DIM / 16) * (SIZE_OUT / 16)) return;
        const int o_base = (gwave & 63) * 16;
        const int b0     = (gwave >> 6) * 16;

        v8f acc = vzero;
        const float* __restrict__ xrow = x + (size_t)(b0 + nl) * SIZE_IN + 2 * kh;
        const float* __restrict__ wrow = weights + (size_t)(o_base + nl) * SIZE_IN + 2 * kh;

        #pragma unroll 4
        for (int k0 = 0; k0 < SIZE_IN; k0 += 4) {
            v2f a, b;
            a.x = xrow[k0];  a.y = xrow[k0 + 1];
            b.x = wrow[k0];  b.y = wrow[k0 + 1];
            acc = wmma_f32_4(a, b, acc);
        }

        const float bo = bias[o_base + nl];
        #pragma unroll
        for (int v = 0; v < 8; ++v) {
            const int m = v + 8 * kh;
            out[(size_t)(b0 + m) * SIZE_OUT + o_base + nl] = acc[v] + bo;
        }
    }
}

extern "C" void kernel_launch(void* const* d_in, const int* in_sizes, int n_in,
                              void* d_out, int out_size, void* d_ws, size_t ws_size,
                              hipStream_t stream) {
    const float* x       = (const float*)d_in[0];
    const float* weights = (const float*)d_in[1];
    const float* bias    = (const float*)d_in[2];
    const float* iw      = (const float*)d_in[3];
    const int*   indices = (const int*)d_in[4];
    const int*   use_idx = (const int*)d_in[5];
    float*       out     = (float*)d_out;

    // Indexed path: 512 blocks x 8 waves (one o_group per block, 16 batch tiles split 2 ways).
    // Plain path uses the first 1024 waves and early-exits the rest.
    dim3 grid(512), block(256);
    hipLaunchKernelGGL(IndexedLinearLayer_wmma_kernel, grid, block, 0, stream,
                       x, weights, bias, iw, indices, use_idx, out);
}